// AttnPainterOilDensity_13443247636830
// MI455X (gfx1250) — compile-verified
//
#include <hip/hip_runtime.h>
#include <stdint.h>

#define TOPK   10
#define ATH    0.1f
#define NSTR   256
#define HW     16384          /* 128*128 */
#define BLK    256
#define CHUNK  16
#define NCH    (NSTR / CHUNK)

// Issue CHUNK async global->LDS copies for this lane's alpha column.
// lane_base = &alpha[b][0][hw_pixel]; stride between strokes is HW floats.
// lds_col   = &buf[pb][0][tid]; stride between rows is BLK floats.
// ASYNCcnt += CHUNK per wave; completion is in-order per wave, and each lane
// reads back only LDS it wrote itself, so s_wait_asynccnt is sufficient (no barrier).
__device__ __forceinline__ void async_alpha_chunk(const float* lane_base, int n0,
                                                  float* lds_col) {
#pragma unroll
  for (int j = 0; j < CHUNK; ++j) {
    uint32_t l = (uint32_t)(uintptr_t)(lds_col + j * BLK);          // low 32 bits = LDS offset
    uint64_t g = (uint64_t)(uintptr_t)(lane_base + (size_t)(n0 + j) * HW);
    asm volatile("global_load_async_to_lds_b32 %0, %1, off"
                 :: "v"(l), "v"(g) : "memory");
  }
}

__global__ __launch_bounds__(BLK) void paint_topk_kernel(
    const float* __restrict__ color_stroke,  // (4,256,3,128,128)
    const float* __restrict__ alpha,         // (4,256,1,128,128)
    const float* __restrict__ params,        // (4,256,8)
    float* __restrict__ out)                 // canvas (4,3,128,128) ++ den (4,1,128,128)
{
  __shared__ float buf[2][CHUNK][BLK];   // 32 KB alpha staging, double buffered
  __shared__ int   tk_i[BLK][TOPK];      // per-pixel top-k stroke indices
  __shared__ float tk_v[BLK][TOPK];      // per-pixel top-k raw alphas

  const int tid = threadIdx.x;
  const int gid = blockIdx.x * BLK + tid;   // pixel id 0..65535 (w-adjacent lanes)
  const int b   = gid >> 14;                // 16384 pixels per batch image
  const int hwp = gid & (HW - 1);

  const float* abase = alpha + (size_t)b * NSTR * HW + hwp;

  int cnt = 0;

  // Prefetch the highest chunk (strokes 240..255) into buffer 0.
  async_alpha_chunk(abase, (NCH - 1) * CHUNK, &buf[0][0][tid]);

  // Descending stroke scan, chunked + double buffered. top_k == highest stroke
  // indices with alpha > thresh, in descending order.
  for (int ci = NCH - 1; ci >= 0; --ci) {
    const int pb = (NCH - 1 - ci) & 1;
    if (ci > 0) {
      async_alpha_chunk(abase, (ci - 1) * CHUNK, &buf[pb ^ 1][0][tid]);
      asm volatile("s_wait_asynccnt 0x10" ::: "memory");  // chunk ci has landed
    } else {
      asm volatile("s_wait_asynccnt 0x0" ::: "memory");
    }
#pragma unroll
    for (int j = CHUNK - 1; j >= 0; --j) {
      float a = buf[pb][j][tid];
      if (cnt < TOPK && a > ATH) {
        tk_i[tid][cnt] = ci * CHUNK + j;
        tk_v[tid][cnt] = a;                 // gather of RAW alpha (reference gathers alpha, not masked)
        ++cnt;
      }
    }
    if (!__any(cnt < TOPK)) break;          // wave32 vote: uniform early exit
  }

  // Rare fill path (<10 strokes above threshold at some lane's pixel):
  // jax top_k breaks ties (draw==0) by ascending index.
  if (__any(cnt < TOPK)) {
    for (int n = 0; n < NSTR; ++n) {
      if (!__any(cnt < TOPK)) break;
      float a = abase[(size_t)n * HW];
      if (cnt < TOPK && a <= ATH) {
        tk_i[tid][cnt] = n;
        tk_v[tid][cnt] = a;
        ++cnt;
      }
    }
  }

  // Pull the 10 slots into registers (static indices -> no scratch).
  int   id[TOPK];
  float av[TOPK];
#pragma unroll
  for (int s = 0; s < TOPK; ++s) { id[s] = tk_i[tid][s]; av[s] = tk_v[tid][s]; }

  const float* cbase = color_stroke + (size_t)b * NSTR * 3 * HW + hwp;

  // Kick the 30 scattered color cachelines before the dependent composite chain.
#pragma unroll
  for (int s = 0; s < TOPK; ++s) {
    const float* cp = cbase + (size_t)id[s] * 3 * HW;
    __builtin_prefetch(cp, 0, 0);           // -> global_prefetch_b8
    __builtin_prefetch(cp + HW, 0, 0);
    __builtin_prefetch(cp + 2 * HW, 0, 0);
  }

  // Back-to-front composite: slot TOPK-1 first, slot 0 on top (matches reference loop).
  float c0 = 1.f, c1 = 1.f, c2 = 1.f, dn = 1.f;
#pragma unroll
  for (int s = TOPK - 1; s >= 0; --s) {
    const int   i  = id[s];
    const float a  = av[s];
    const float2 p = *(const float2*)(params + ((size_t)b * NSTR + i) * 8 + 2);
    const float sv = (a > ATH) ? p.x * p.y : 0.f;   // stroke_s masked by (alpha>thresh)
    const float* cp = cbase + (size_t)i * 3 * HW;
    const float r = cp[0], g = cp[HW], bl = cp[2 * HW];
    const float om = 1.f - a;
    c0 = c0 * om + a * r;
    c1 = c1 * om + a * g;
    c2 = c2 * om + a * bl;
    dn = dn * om + a * sv;
  }

  float* canv = out + (size_t)b * 3 * HW + hwp;     // canvas (B,3,H,W)
  canv[0]      = c0;
  canv[HW]     = c1;
  canv[2 * HW] = c2;
  out[(size_t)4 * 3 * HW + (size_t)b * HW + hwp] = dn;  // den_map appended flat
}

extern "C" void kernel_launch(void* const* d_in, const int* in_sizes, int n_in,
                              void* d_out, int out_size, void* d_ws, size_t ws_size,
                              hipStream_t stream) {
  (void)in_sizes; (void)n_in; (void)out_size; (void)d_ws; (void)ws_size;
  const float* color_stroke = (const float*)d_in[0];   // (4,256,3,128,128) f32
  const float* alpha        = (const float*)d_in[1];   // (4,256,1,128,128) f32
  const float* params       = (const float*)d_in[2];   // (4,256,8) f32
  float*       out          = (float*)d_out;           // 196608 + 65536 f32

  const int pixels = 4 * HW;                 // 65536
  dim3 grid(pixels / BLK), block(BLK);
  paint_topk_kernel<<<grid, block, 0, stream>>>(color_stroke, alpha, params, out);
}